// model_25933012533459
// MI455X (gfx1250) — compile-verified
//
#include <hip/hip_runtime.h>
#include <stdint.h>

#define BATCH   16
#define CHANS   80
#define KTOP    100
#define NSCALE  5
#define NBINS   4096
#define CAP     4096
#define FINAL_K 300
#define DETS_PER_B (NSCALE * KTOP)   // 500
#define ADEPTH  4                    // async pipeline depth (LDS buffers)

// ---------------------------------------------------------------------------
// helpers
// ---------------------------------------------------------------------------
__device__ __forceinline__ unsigned fkey(float f) {
    unsigned u = __float_as_uint(f);
    return (u & 0x80000000u) ? ~u : (u | 0x80000000u);   // monotone order-preserving key
}

// in-LDS bitonic sort, descending, n = power of two, whole block cooperates
__device__ inline void bitonic_desc(unsigned long long* a, int n) {
    for (int k = 2; k <= n; k <<= 1) {
        for (int j = k >> 1; j > 0; j >>= 1) {
            for (int i = threadIdx.x; i < n; i += blockDim.x) {
                int ix = i ^ j;
                if (ix > i) {
                    unsigned long long x = a[i], y = a[ix];
                    bool descBlk = ((i & k) == 0);
                    bool sw = descBlk ? (x < y) : (x > y);
                    if (sw) { a[i] = y; a[ix] = x; }
                }
            }
            __syncthreads();
        }
    }
}

// ---------------------------------------------------------------------------
// 0) zero the histogram / threshold / counter region of the workspace
// ---------------------------------------------------------------------------
__global__ void zero_kernel(unsigned* p, int n) {
    int i = blockIdx.x * blockDim.x + threadIdx.x;
    if (i < n) p[i] = 0u;
}

// ---------------------------------------------------------------------------
// 1) histogram of heat values, streamed through LDS with CDNA5 async copies.
//    4-deep pipeline; every lane always issues (index clamped to V-1) so the
//    per-wave ASYNCcnt sequence is statically known and `s_wait_asynccnt 0x3`
//    always releases exactly the tile being consumed.
//    blockIdx.y = batch. V = number of float4 vectors per batch image.
// ---------------------------------------------------------------------------
__global__ void hist_kernel(const float* __restrict__ heat, long long V,
                            int scale, unsigned* __restrict__ hist)
{
    __shared__ unsigned lhist[NBINS];             // 16 KB
    __shared__ float    buf[ADEPTH][256 * 4];     // 16 KB: 4 buffers x (256 lanes x 16B)

    const int tid  = threadIdx.x;
    const int b    = blockIdx.y;
    const int pair = b * NSCALE + scale;

    for (int i = tid; i < NBINS; i += blockDim.x) lhist[i] = 0u;
    __syncthreads();

    const float* hb = heat + (long long)b * V * 4;
    const unsigned lds0 = (unsigned)(size_t)(&buf[0][0]);   // low 32 bits = LDS byte offset

    const long long first  = (long long)blockIdx.x * 256;   // launcher guarantees first < V
    const long long stride = (long long)gridDim.x * 256;

    // issue helper: ALWAYS issues; clamps the vector index into range so that
    // every wave executes an identical async-instruction sequence.
    auto issue = [&](long long t, int bsel) {
        long long v = t + tid;
        if (v > V - 1) v = V - 1;                 // duplicate tail load, discarded later
        const float* g = hb + v * 4;
        unsigned lo = lds0 + (unsigned)bsel * 4096u + (unsigned)tid * 16u;
        asm volatile("global_load_async_to_lds_b128 %0, %1, off"
                     :: "v"(lo), "v"(g) : "memory");
    };

    // prologue: fill pipeline with ADEPTH-1 tiles
    issue(first + 0 * stride, 0);
    issue(first + 1 * stride, 1);
    issue(first + 2 * stride, 2);

    int sel = 0;
    for (long long it = first; it < V; it += stride) {
        issue(it + (ADEPTH - 1) * stride, (sel + ADEPTH - 1) & (ADEPTH - 1));
        // oldest of the (up to 4) outstanding copies is the tile we consume now
        asm volatile("s_wait_asynccnt 0x3" ::: "memory");

        long long cv = it + tid;
        if (cv < V) {
            float4 x = *(const float4*)&buf[sel][tid * 4];
            float vv[4] = { x.x, x.y, x.z, x.w };
#pragma unroll
            for (int j = 0; j < 4; j++) {
                int bin = (int)(vv[j] * (float)NBINS);
                bin = bin < 0 ? 0 : (bin > NBINS - 1 ? NBINS - 1 : bin);
                atomicAdd(&lhist[bin], 1u);
            }
        }
        sel = (sel + 1) & (ADEPTH - 1);
    }
    // remaining clamped-duplicate copies drain at s_endpgm (implicit wait-idle)

    __syncthreads();
    unsigned* gh = hist + (long long)pair * NBINS;
    for (int i = tid; i < NBINS; i += blockDim.x)
        if (lhist[i]) atomicAdd(&gh[i], lhist[i]);
}

// ---------------------------------------------------------------------------
// 2) find per-(batch,scale) bin threshold containing the KTOP-th value
// ---------------------------------------------------------------------------
__global__ void select_kernel(const unsigned* __restrict__ hist, unsigned* __restrict__ thresh)
{
    __shared__ unsigned sh[NBINS];
    const int pair = blockIdx.x;
    const unsigned* h = hist + (long long)pair * NBINS;
    for (int i = threadIdx.x; i < NBINS; i += blockDim.x) sh[i] = h[i];
    __syncthreads();
    if (threadIdx.x == 0) {
        unsigned acc = 0; unsigned t = 0;
        for (int bin = NBINS - 1; bin >= 0; --bin) {
            acc += sh[bin];
            if (acc >= (unsigned)KTOP) { t = (unsigned)bin; break; }
        }
        thresh[pair] = t;
    }
}

// ---------------------------------------------------------------------------
// 3) collect candidates >= threshold (second pass is L2-resident: whole heat
//    working set (~175MB) fits the 192MB L2)
// ---------------------------------------------------------------------------
__global__ void collect_kernel(const float* __restrict__ heat, long long V, int scale,
                               const unsigned* __restrict__ thresh,
                               unsigned* __restrict__ counts, uint2* __restrict__ cand)
{
    const int b    = blockIdx.y;
    const int pair = b * NSCALE + scale;
    const unsigned tb = thresh[pair];
    const float* hb = heat + (long long)b * V * 4;
    uint2* cp = cand + (long long)pair * CAP;

    const long long stride = (long long)gridDim.x * blockDim.x;
    for (long long v = (long long)blockIdx.x * blockDim.x + threadIdx.x; v < V; v += stride) {
        float4 x = *(const float4*)(hb + v * 4);
        float vv[4] = { x.x, x.y, x.z, x.w };
        float sel[4]; unsigned si[4]; int c = 0;
#pragma unroll
        for (int j = 0; j < 4; j++) {
            int bin = (int)(vv[j] * (float)NBINS);
            bin = bin < 0 ? 0 : (bin > NBINS - 1 ? NBINS - 1 : bin);
            if ((unsigned)bin >= tb) { sel[c] = vv[j]; si[c] = (unsigned)(v * 4 + j); c++; }
        }
        if (c) {
            unsigned pos = atomicAdd(&counts[pair], (unsigned)c);
            for (int i = 0; i < c; i++) {
                unsigned p = pos + i;
                if (p < CAP) cp[p] = make_uint2(__float_as_uint(sel[i]), si[i]);
            }
        }
    }
}

// ---------------------------------------------------------------------------
// 4) per (batch,scale): sort candidates, take top-100, decode boxes,
//    invalidate, re-sort 100, write dets(b, 500, 8)
// ---------------------------------------------------------------------------
__global__ void scale_topk_kernel(const unsigned* __restrict__ counts,
                                  const uint2* __restrict__ cand,
                                  const float* tl0, const float* tl1, const float* tl2,
                                  const float* tl3, const float* tl4,
                                  const float* br0, const float* br1, const float* br2,
                                  const float* br3, const float* br4,
                                  float* __restrict__ dets)
{
    __shared__ unsigned long long keys[CAP];     // 32 KB
    __shared__ float stage[KTOP][6];
    __shared__ unsigned long long k2[128];

    const int pair = blockIdx.x;
    const int b = pair / NSCALE;
    const int s = pair - b * NSCALE;
    const int W = 160 >> s;
    const int HW = W * W;

    const float* tlp; const float* brp;
    switch (s) {
        case 0:  tlp = tl0; brp = br0; break;
        case 1:  tlp = tl1; brp = br1; break;
        case 2:  tlp = tl2; brp = br2; break;
        case 3:  tlp = tl3; brp = br3; break;
        default: tlp = tl4; brp = br4; break;
    }
    tlp += (long long)b * 2 * HW;
    brp += (long long)b * 2 * HW;

    unsigned cnt = counts[pair]; if (cnt > CAP) cnt = CAP;
    const uint2* cp = cand + (long long)pair * CAP;

    for (int i = threadIdx.x; i < CAP; i += blockDim.x) {
        if (i < (int)cnt) {
            uint2 e = cp[i];
            keys[i] = ((unsigned long long)fkey(__uint_as_float(e.x)) << 32) | (unsigned)i;
        } else {
            keys[i] = 0ull;
        }
    }
    __syncthreads();
    bitonic_desc(keys, CAP);

    if (threadIdx.x < KTOP) {
        int j = threadIdx.x;
        unsigned pos = (unsigned)keys[j];
        uint2 e = cp[pos];
        float val = __uint_as_float(e.x);
        unsigned idx = e.y;                  // flat index in [0, C*H*W)
        int cls = (int)(idx / (unsigned)HW);
        int ihw = (int)(idx - (unsigned)cls * (unsigned)HW);
        int y = ihw / W;
        int x = ihw - y * W;
        // BASE_LAYER_RANGE/8/2 -> min=1.5, max=3.0 ; (max-min)=1.5, (max+min)/2=2.25
        float t0 = tlp[ihw], t1 = tlp[HW + ihw];
        float r0 = brp[ihw], r1 = brp[HW + ihw];
        float tlx = (float)x - (1.5f * t0 + 2.25f);
        float tly = (float)y - (1.5f * t1 + 2.25f);
        float brx = (float)x + (1.5f * r0 + 2.25f);
        float bry = (float)y + (1.5f * r1 + 2.25f);
        float sc  = ((brx < tlx) || (bry < tly)) ? -1.0f : val;
        stage[j][0] = tlx; stage[j][1] = tly; stage[j][2] = brx; stage[j][3] = bry;
        stage[j][4] = sc;  stage[j][5] = (float)cls;
        k2[j] = ((unsigned long long)fkey(sc) << 32) | (unsigned)(0xFFFFFFFFu - (unsigned)j);
    }
    if (threadIdx.x >= KTOP && threadIdx.x < 128) k2[threadIdx.x] = 0ull;
    __syncthreads();
    bitonic_desc(k2, 128);

    if (threadIdx.x < KTOP) {
        int r = threadIdx.x;
        int j = (int)(0xFFFFFFFFu - (unsigned)k2[r]);
        float wsc = 160.0f / (float)W;       // width_scale == height_scale (square maps)
        float* d = dets + ((long long)b * DETS_PER_B + s * KTOP + r) * 8;
        d[0] = stage[j][0] * wsc;
        d[1] = stage[j][1] * wsc;
        d[2] = stage[j][2] * wsc;
        d[3] = stage[j][3] * wsc;
        d[4] = stage[j][4];
        d[5] = stage[j][4];
        d[6] = stage[j][4];
        d[7] = stage[j][5];
    }
}

// ---------------------------------------------------------------------------
// 5) per batch: top-300 of 500 dets by score, gather rows to output
// ---------------------------------------------------------------------------
__global__ void final_kernel(const float* __restrict__ dets, float* __restrict__ out)
{
    __shared__ unsigned long long keys[512];
    const int b = blockIdx.x;
    const float* db = dets + (long long)b * DETS_PER_B * 8;

    for (int i = threadIdx.x; i < 512; i += blockDim.x) {
        if (i < DETS_PER_B)
            keys[i] = ((unsigned long long)fkey(db[i * 8 + 4]) << 32)
                      | (unsigned)(0xFFFFFFFFu - (unsigned)i);
        else
            keys[i] = 0ull;
    }
    __syncthreads();
    bitonic_desc(keys, 512);

    for (int r = threadIdx.x; r < FINAL_K; r += blockDim.x) {
        int i = (int)(0xFFFFFFFFu - (unsigned)keys[r]);
        const float* srcp = db + (long long)i * 8;
        float* dst = out + ((long long)b * FINAL_K + r) * 8;
#pragma unroll
        for (int t = 0; t < 8; t++) dst[t] = srcp[t];
    }
}

// ---------------------------------------------------------------------------
// host-side launcher
// ---------------------------------------------------------------------------
extern "C" void kernel_launch(void* const* d_in, const int* in_sizes, int n_in,
                              void* d_out, int out_size, void* d_ws, size_t ws_size,
                              hipStream_t stream)
{
    (void)in_sizes; (void)n_in; (void)out_size; (void)ws_size;

    // input order: heat0, tl0, br0, heat1, tl1, br1, ...
    const float* heat[NSCALE]; const float* tl[NSCALE]; const float* br[NSCALE];
    for (int s = 0; s < NSCALE; s++) {
        heat[s] = (const float*)d_in[3 * s + 0];
        tl[s]   = (const float*)d_in[3 * s + 1];
        br[s]   = (const float*)d_in[3 * s + 2];
    }

    // workspace layout
    char* ws = (char*)d_ws;
    const size_t histBytes = (size_t)BATCH * NSCALE * NBINS * sizeof(unsigned); // 1,310,720
    unsigned* hist   = (unsigned*)ws;
    unsigned* thresh = (unsigned*)(ws + histBytes);             // 80 u32
    unsigned* counts = (unsigned*)(ws + histBytes + 512);       // 80 u32
    uint2*    cand   = (uint2*)   (ws + histBytes + 1024);      // 80*CAP*8 B
    float*    dets   = (float*)   (ws + histBytes + 1024 + (size_t)BATCH * NSCALE * CAP * 8);

    // zero hist + thresh + counts (ws is poisoned; graph replays don't re-zero)
    int zn = BATCH * NSCALE * NBINS + 256;
    zero_kernel<<<dim3((zn + 255) / 256), 256, 0, stream>>>(hist, zn);

    static const int HSs[NSCALE] = { 160, 80, 40, 20, 10 };

    for (int s = 0; s < NSCALE; s++) {
        long long N = (long long)CHANS * HSs[s] * HSs[s];
        long long V = N / 4;                            // all sizes divisible by 4
        long long gx = (V + 255) / 256; if (gx > 128) gx = 128; if (gx < 1) gx = 1;
        hist_kernel<<<dim3((unsigned)gx, BATCH), 256, 0, stream>>>(heat[s], V, s, hist);
    }

    select_kernel<<<dim3(BATCH * NSCALE), 256, 0, stream>>>(hist, thresh);

    for (int s = 0; s < NSCALE; s++) {
        long long V = (long long)CHANS * HSs[s] * HSs[s] / 4;
        long long gx = (V + 1023) / 1024; if (gx > 128) gx = 128; if (gx < 1) gx = 1;
        collect_kernel<<<dim3((unsigned)gx, BATCH), 256, 0, stream>>>(heat[s], V, s,
                                                                      thresh, counts, cand);
    }

    scale_topk_kernel<<<dim3(BATCH * NSCALE), 256, 0, stream>>>(
        counts, cand,
        tl[0], tl[1], tl[2], tl[3], tl[4],
        br[0], br[1], br[2], br[3], br[4],
        dets);

    final_kernel<<<dim3(BATCH), 256, 0, stream>>>(dets, (float*)d_out);
}